// GCNSubModule_1451698946200
// MI455X (gfx1250) — compile-verified
//
#include <hip/hip_runtime.h>

#define D 128               // D_IN == D_OUT == 128
#define BN_EPS 1e-5f

typedef float v2f __attribute__((ext_vector_type(2)));
typedef float v8f __attribute__((ext_vector_type(8)));

// ---------------------------------------------------------------------------
// 1) deg[i] = 1.0 (self-loop) ; zero the 256 floats of BN stat accumulators
// ---------------------------------------------------------------------------
__global__ void init_deg_kernel(float* __restrict__ deg, float* __restrict__ stats,
                                int N) {
    int i = blockIdx.x * blockDim.x + threadIdx.x;
    if (i < N) deg[i] = 1.0f;           // self-loop contributes 1 to every degree
    if (i < 256) stats[i] = 0.0f;       // sums[128] + sumsq[128]
}

// ---------------------------------------------------------------------------
// 2) deg[dst[e]] += 1 over all edges
// ---------------------------------------------------------------------------
__global__ void degree_kernel(const int* __restrict__ dst, float* __restrict__ deg,
                              int E) {
    int e = blockIdx.x * blockDim.x + threadIdx.x;
    if (e < E) unsafeAtomicAdd(&deg[dst[e]], 1.0f);
}

// ---------------------------------------------------------------------------
// 3) dinv[i] = rsqrt(deg[i])   (deg >= 1 always, self-loops included)
// ---------------------------------------------------------------------------
__global__ void rsqrt_kernel(float* __restrict__ deg, int N) {
    int i = blockIdx.x * blockDim.x + threadIdx.x;
    if (i < N) deg[i] = rsqrtf(deg[i]);
}

// ---------------------------------------------------------------------------
// 4) xw = x @ W via V_WMMA_F32_16X16X4_F32 (full fp32 precision).
//    One block = 16 rows of x staged in LDS; 8 waves cover the 8 column tiles.
//    K=128 -> 32 WMMA steps of K=4 per wave.
//    A layout (ISA 7.12.2, 32-bit A 16x4): lane L (0..15) row=L, VGPR0=K0,
//    VGPR1=K1; lanes 16..31 row=L-16, K2/K3.  B 4x16 mirrors it transposed.
// ---------------------------------------------------------------------------
__global__ __launch_bounds__(256) void gemm_wmma_kernel(
        const float* __restrict__ x, const float* __restrict__ W,
        float* __restrict__ xw, int N) {
    __shared__ float xs[16 * D];                       // 8 KB tile of x
    const int tid  = threadIdx.x;
    const int m0   = blockIdx.x * 16;

    // stage 16x128 floats = 512 float4 with 256 threads (2 each)
    const float4* xg  = (const float4*)(x + (size_t)m0 * D);
    float4*       xs4 = (float4*)xs;
    xs4[tid]       = xg[tid];
    xs4[tid + 256] = xg[tid + 256];
    __syncthreads();

    const int wave  = tid >> 5;
    const int lane  = tid & 31;
    const int n0    = wave * 16;          // this wave's 16-column tile
    const int l15   = lane & 15;
    const int khalf = (lane >> 4) * 2;    // lanes 16..31 hold K+2 / K+3

    v8f acc = {};
#pragma unroll 8
    for (int k = 0; k < D; k += 4) {
        const int kb = k + khalf;                       // even -> 8B aligned
        v2f a = *(const v2f*)&xs[l15 * D + kb];         // A: row l15, K=kb,kb+1
        v2f b;
        b.x = W[(kb + 0) * D + n0 + l15];               // B: K=kb,   N=n0+l15
        b.y = W[(kb + 1) * D + n0 + l15];               // B: K=kb+1, N=n0+l15
        acc = __builtin_amdgcn_wmma_f32_16x16x4_f32(
                false, a, false, b, (short)0, acc, false, false);
    }

    // D layout: VGPR v -> row m0+v (lanes 0..15) / m0+8+v (lanes 16..31)
    const int rbase = m0 + ((lane >> 4) << 3);
    const int col   = n0 + l15;
#pragma unroll
    for (int v = 0; v < 8; ++v)
        xw[(size_t)(rbase + v) * D + col] = acc[v];
}

// ---------------------------------------------------------------------------
// 5) out = self-loop contribution + bias:  out[i,c] = xw[i,c]*dinv[i]^2 + b[c]
// ---------------------------------------------------------------------------
__global__ void selfloop_kernel(const float* __restrict__ xw,
                                const float* __restrict__ dinv,
                                const float* __restrict__ b,
                                float* __restrict__ out, int N) {
    size_t t = (size_t)blockIdx.x * blockDim.x + threadIdx.x;
    if (t < (size_t)N * D) {
        int row = (int)(t >> 7);
        int col = (int)(t & (D - 1));
        float di = dinv[row];
        out[t] = xw[t] * di * di + b[col];
    }
}

// ---------------------------------------------------------------------------
// 6) edge scatter: one wave per edge, each lane handles 4 channels (float4).
//    out[dst] += xw[src] * dinv[src]*dinv[dst]
// ---------------------------------------------------------------------------
__global__ void scatter_kernel(const int* __restrict__ src,
                               const int* __restrict__ dst,
                               const float* __restrict__ dinv,
                               const float* __restrict__ xw,
                               float* __restrict__ out, int E) {
    long long t = (long long)blockIdx.x * blockDim.x + threadIdx.x;
    int e = (int)(t >> 5);                 // uniform within a wave32
    if (e >= E) return;
    int lane = (int)(t & 31);
    int s = src[e];
    int d = dst[e];
    float w = dinv[s] * dinv[d];
    const float4 m = ((const float4*)(xw + (size_t)s * D))[lane];
    float* o = out + (size_t)d * D + lane * 4;
    unsafeAtomicAdd(o + 0, m.x * w);
    unsafeAtomicAdd(o + 1, m.y * w);
    unsafeAtomicAdd(o + 2, m.z * w);
    unsafeAtomicAdd(o + 3, m.w * w);
}

// ---------------------------------------------------------------------------
// 7) BN statistics: per-column sum and sum-of-squares over the node dim.
// ---------------------------------------------------------------------------
__global__ void bnstats_kernel(const float* __restrict__ out,
                               float* __restrict__ sums,
                               float* __restrict__ sumsq, int N) {
    int t      = blockIdx.x * blockDim.x + threadIdx.x;
    int col    = t & (D - 1);
    int row    = t >> 7;
    int stride = (gridDim.x * blockDim.x) >> 7;
    float s = 0.0f, q = 0.0f;
    for (int r = row; r < N; r += stride) {
        float v = out[(size_t)r * D + col];
        s += v;
        q += v * v;
    }
    unsafeAtomicAdd(&sums[col], s);
    unsafeAtomicAdd(&sumsq[col], q);
}

// ---------------------------------------------------------------------------
// 8) BN finalize + ReLU, in place on d_out.
// ---------------------------------------------------------------------------
__global__ void bnfinal_kernel(float* __restrict__ out,
                               const float* __restrict__ sums,
                               const float* __restrict__ sumsq,
                               const float* __restrict__ gamma,
                               const float* __restrict__ beta, int N) {
    size_t t = (size_t)blockIdx.x * blockDim.x + threadIdx.x;
    if (t >= (size_t)N * D) return;
    int col = (int)(t & (D - 1));
    float invN = 1.0f / (float)N;
    float mean = sums[col] * invN;
    float var  = sumsq[col] * invN - mean * mean;
    float v = (out[t] - mean) * rsqrtf(var + BN_EPS) * gamma[col] + beta[col];
    out[t] = fmaxf(v, 0.0f);
}

// ---------------------------------------------------------------------------
extern "C" void kernel_launch(void* const* d_in, const int* in_sizes, int n_in,
                              void* d_out, int out_size, void* d_ws, size_t ws_size,
                              hipStream_t stream) {
    const float* x     = (const float*)d_in[0];
    const int*   ei    = (const int*)d_in[1];    // [2, E] (harness int contract)
    const float* W     = (const float*)d_in[2];
    const float* b     = (const float*)d_in[3];
    const float* gamma = (const float*)d_in[4];
    const float* beta  = (const float*)d_in[5];
    float*       out   = (float*)d_out;

    const int N = in_sizes[0] / D;               // 100000 (== 6250 * 16)
    const int E = in_sizes[1] / 2;               // 3200000
    const int* src = ei;
    const int* dst = ei + E;

    // workspace layout (floats): xw[N*D] | deg/dinv[N] | sums[128] | sumsq[128]
    float* ws    = (float*)d_ws;
    float* xw    = ws;
    float* deg   = xw + (size_t)N * D;
    float* sums  = deg + N;
    float* sumsq = sums + D;

    const int T = 256;
    const size_t ND = (size_t)N * D;

    init_deg_kernel<<<(N + T - 1) / T, T, 0, stream>>>(deg, sums, N);
    degree_kernel<<<(E + T - 1) / T, T, 0, stream>>>(dst, deg, E);
    rsqrt_kernel<<<(N + T - 1) / T, T, 0, stream>>>(deg, N);
    gemm_wmma_kernel<<<N / 16, T, 0, stream>>>(x, W, xw, N);
    selfloop_kernel<<<(int)((ND + T - 1) / T), T, 0, stream>>>(xw, deg, b, out, N);
    {
        long long threads = (long long)E * 32;
        int blocks = (int)((threads + T - 1) / T);
        scatter_kernel<<<blocks, T, 0, stream>>>(src, dst, deg, xw, out, E);
    }
    bnstats_kernel<<<1024, T, 0, stream>>>(out, sums, sumsq, N);
    bnfinal_kernel<<<(int)((ND + T - 1) / T), T, 0, stream>>>(out, sums, sumsq,
                                                              gamma, beta, N);
}